// LogEuclideanLoss_72181220376965
// MI455X (gfx1250) — compile-verified
//
#include <hip/hip_runtime.h>
#include <hip/hip_bf16.h>

typedef __attribute__((ext_vector_type(2))) float v2f;
typedef __attribute__((ext_vector_type(8))) float v8f;

#define LE_EPS 1e-10f

// Sanitize like jnp.nan_to_num(nan=eps, posinf=eps, neginf=eps) followed by abs().
__device__ __forceinline__ float le_sanitize(float x) {
    float a = fabsf(x);
    // NaN fails the compare (-> eps); +/-inf fails (-> eps); else abs value.
    return (a < 1e37f) ? a : LE_EPS;
}

// Closed-form log of a 3x3 SPD matrix (symmetric input, 6 unique entries).
// Eigenvalues via the trigonometric method; log(A) evaluated as the Newton
// divided-difference quadratic in A (stable for clustered eigenvalues):
//   log(A) = f1*I + f[w1,w2]*(A - w1 I) + f[w1,w2,w3]*(A^2 - (w1+w2)A + w1 w2 I)
__device__ __forceinline__ void log_spd3(
    float a00, float a01, float a02, float a11, float a12, float a22,
    float& l00, float& l01, float& l02, float& l11, float& l12, float& l22)
{
    const float third = 1.0f / 3.0f;
    float q  = (a00 + a11 + a22) * third;
    float d0 = a00 - q, d1 = a11 - q, d2 = a22 - q;
    float p1 = a01 * a01 + a02 * a02 + a12 * a12;
    float p2 = d0 * d0 + d1 * d1 + d2 * d2 + 2.0f * p1;
    float p  = sqrtf(p2 * (1.0f / 6.0f)) + 1e-30f;
    float ip = 1.0f / p;

    // B = (A - q I) / p ; r = det(B)/2 in [-1,1]
    float b00 = d0 * ip, b11 = d1 * ip, b22 = d2 * ip;
    float b01 = a01 * ip, b02 = a02 * ip, b12 = a12 * ip;
    float detB = b00 * (b11 * b22 - b12 * b12)
               - b01 * (b01 * b22 - b12 * b02)
               + b02 * (b01 * b12 - b11 * b02);
    float r = fminf(1.0f, fmaxf(-1.0f, 0.5f * detB));

    float phi = acosf(r) * third;
    float tp  = 2.0f * p;
    float w1  = q + tp * __cosf(phi);                        // largest
    float w3  = q + tp * __cosf(phi + 2.0943951023931953f);  // smallest
    float w2  = 3.0f * q - w1 - w3;                          // middle

    // f(w) = log(w + eps) + eps  (matches reference lw = log(w+EPS)+EPS)
    float f1 = __logf(w1 + LE_EPS) + LE_EPS;
    float f2 = __logf(w2 + LE_EPS) + LE_EPS;
    float f3 = __logf(w3 + LE_EPS) + LE_EPS;

    // First divided differences with derivative fallback (f' = 1/(w+eps)).
    float d12 = w1 - w2, d23 = w2 - w3, d13 = w1 - w3;
    float g12 = (d12 > 1e-6f * (w1 + w2)) ? (f1 - f2) / d12
                                          : 1.0f / (0.5f * (w1 + w2) + LE_EPS);
    float g23 = (d23 > 1e-6f * (w2 + w3)) ? (f2 - f3) / d23
                                          : 1.0f / (0.5f * (w2 + w3) + LE_EPS);
    // Second divided difference; limit is f''(w2)/2 = -1/(2 (w2+eps)^2).
    float m2 = w2 + LE_EPS;
    float h  = (d13 > 1e-6f * (w1 + w3)) ? (g12 - g23) / d13
                                         : -0.5f / (m2 * m2);

    // S = A^2 (symmetric)
    float s00 = a00 * a00 + a01 * a01 + a02 * a02;
    float s01 = a00 * a01 + a01 * a11 + a02 * a12;
    float s02 = a00 * a02 + a01 * a12 + a02 * a22;
    float s11 = a01 * a01 + a11 * a11 + a12 * a12;
    float s12 = a01 * a02 + a11 * a12 + a12 * a22;
    float s22 = a02 * a02 + a12 * a12 + a22 * a22;

    float ws = w1 + w2;       // (w1 + w2)
    float wp = w1 * w2;       // w1 * w2

    l01 = g12 * a01 + h * (s01 - ws * a01);
    l02 = g12 * a02 + h * (s02 - ws * a02);
    l12 = g12 * a12 + h * (s12 - ws * a12);
    l00 = f1 + g12 * (a00 - w1) + h * (s00 - ws * a00 + wp);
    l11 = f1 + g12 * (a11 - w1) + h * (s11 - ws * a11 + wp);
    l22 = f1 + g12 * (a22 - w1) + h * (s22 - ws * a22 + wp);
}

// Stage 1: one thread per 3x3 matrix pair. Per-lane squared-diff sum, then a
// wave-level reduction using V_WMMA_F32_16X16X4_F32 (B = ones), LDS combine,
// one partial per block into d_ws.
__global__ void __launch_bounds__(256)
log_euclid_stage1(const float* __restrict__ D1, const float* __restrict__ D2,
                  float* __restrict__ partial, int n_mat)
{
    int idx = blockIdx.x * 256 + threadIdx.x;
    bool valid = idx < n_mat;
    int mi = valid ? idx : 0;                 // keep all lanes active (EXEC==all 1s)
    const float* p1 = D1 + (size_t)mi * 9;
    const float* p2 = D2 + (size_t)mi * 9;

    // Unique symmetric entries at row-major offsets 0,1,2,4,5,8 (single-use
    // data -> nontemporal to spare L2 for other traffic).
    float a00 = le_sanitize(__builtin_nontemporal_load(p1 + 0));
    float a01 = le_sanitize(__builtin_nontemporal_load(p1 + 1));
    float a02 = le_sanitize(__builtin_nontemporal_load(p1 + 2));
    float a11 = le_sanitize(__builtin_nontemporal_load(p1 + 4));
    float a12 = le_sanitize(__builtin_nontemporal_load(p1 + 5));
    float a22 = le_sanitize(__builtin_nontemporal_load(p1 + 8));

    float c00 = le_sanitize(__builtin_nontemporal_load(p2 + 0));
    float c01 = le_sanitize(__builtin_nontemporal_load(p2 + 1));
    float c02 = le_sanitize(__builtin_nontemporal_load(p2 + 2));
    float c11 = le_sanitize(__builtin_nontemporal_load(p2 + 4));
    float c12 = le_sanitize(__builtin_nontemporal_load(p2 + 5));
    float c22 = le_sanitize(__builtin_nontemporal_load(p2 + 8));

    float x00, x01, x02, x11, x12, x22;
    float y00, y01, y02, y11, y12, y22;
    log_spd3(a00, a01, a02, a11, a12, a22, x00, x01, x02, x11, x12, x22);
    log_spd3(c00, c01, c02, c11, c12, c22, y00, y01, y02, y11, y12, y22);

    float e00 = x00 - y00, e01 = x01 - y01, e02 = x02 - y02;
    float e11 = x11 - y11, e12 = x12 - y12, e22 = x22 - y22;
    // Full-matrix Frobenius: off-diagonals appear twice.
    float acc = e00 * e00 + e11 * e11 + e22 * e22
              + 2.0f * (e01 * e01 + e02 * e02 + e12 * e12);
    if (!valid) acc = 0.0f;

    // ---- wave32 reduction on the matrix unit ----
    // A (16x4 f32): lane m -> A[m][0]=acc, A[m][1]=0; lane m+16 -> A[m][2]=acc, A[m][3]=0
    // B = ones => D[m][n] = acc[m] + acc[m+16] (independent of n).
    v2f a; a[0] = acc; a[1] = 0.0f;
    v2f b; b[0] = 1.0f; b[1] = 1.0f;
    v8f cz = {};
    v8f d = __builtin_amdgcn_wmma_f32_16x16x4_f32(
        false, a, false, b, (short)0, cz, false, false);
    float s = d[0] + d[1] + d[2] + d[3] + d[4] + d[5] + d[6] + d[7];
    // lanes 0-15 hold sum of rows 0..7, lanes 16-31 hold sum of rows 8..15
    s += __shfl_xor(s, 16, 32);               // full 32-lane sum in every lane

    __shared__ float lds[8];                  // 256 threads = 8 waves
    int wave = threadIdx.x >> 5;
    if ((threadIdx.x & 31) == 0) lds[wave] = s;
    __syncthreads();
    if (threadIdx.x == 0) {
        float t = 0.0f;
        #pragma unroll
        for (int i = 0; i < 8; ++i) t += lds[i];
        partial[blockIdx.x] = t;
    }
}

// Stage 2: deterministic single-block reduction of per-block partials.
__global__ void __launch_bounds__(256)
log_euclid_stage2(const float* __restrict__ partial, int n,
                  float* __restrict__ out, float inv_total)
{
    float s = 0.0f;
    for (int i = threadIdx.x; i < n; i += 256) s += partial[i];
    __shared__ float lds[256];
    lds[threadIdx.x] = s;
    __syncthreads();
    #pragma unroll
    for (int off = 128; off > 0; off >>= 1) {
        if (threadIdx.x < off) lds[threadIdx.x] += lds[threadIdx.x + off];
        __syncthreads();
    }
    if (threadIdx.x == 0) out[0] = lds[0] * inv_total;
}

extern "C" void kernel_launch(void* const* d_in, const int* in_sizes, int n_in,
                              void* d_out, int out_size, void* d_ws, size_t ws_size,
                              hipStream_t stream) {
    const float* D1 = (const float*)d_in[0];
    const float* D2 = (const float*)d_in[1];
    float* out = (float*)d_out;
    float* partial = (float*)d_ws;

    int total_elems = in_sizes[0];            // 18,874,368 (mean denominator)
    int n_mat = total_elems / 9;              // 2,097,152 matrices
    int blocks = (n_mat + 255) / 256;         // 8192 (exact, so EXEC stays full)

    log_euclid_stage1<<<blocks, 256, 0, stream>>>(D1, D2, partial, n_mat);
    log_euclid_stage2<<<1, 256, 0, stream>>>(partial, blocks, out,
                                             1.0f / (float)total_elems);
}